// SpGAT_31353261261497
// MI455X (gfx1250) — compile-verified
//
#include <hip/hip_runtime.h>
#include <hip/hip_bf16.h>

#define NHEADS  4
#define N_NODES 50000
#define N_EDGES 1600000
#define NFEAT   256
#define DH      64          // both DH_IN and DH_OUT are 64
#define ALPHA   0.2f

typedef float v2f __attribute__((ext_vector_type(2)));
typedef float v8f __attribute__((ext_vector_type(8)));

// ---------------------------------------------------------------------------
// Kernel 1: fold attention vector into W:  wl = W @ a_l, wr = W @ a_r  (64 ea)
// ---------------------------------------------------------------------------
__global__ void gat_prep(const float* __restrict__ W, const float* __restrict__ a,
                         float* __restrict__ wl, float* __restrict__ wr) {
    int k = threadIdx.x;
    if (k < DH) {
        float s1 = 0.f, s2 = 0.f;
        for (int j = 0; j < DH; ++j) {
            float w = W[k * DH + j];
            s1 += w * a[j];
            s2 += w * a[DH + j];
        }
        wl[k] = s1;
        wr[k] = s2;
    }
}

// ---------------------------------------------------------------------------
// Kernel 2: zero the accumulators (d_out is used as agg; rowsum in ws)
// ---------------------------------------------------------------------------
__global__ void gat_zero(float* __restrict__ out, float* __restrict__ rowsum) {
    long long i = (long long)blockIdx.x * 256 + threadIdx.x;
    if (i < (long long)N_NODES * NFEAT) out[i] = 0.f;
    if (i < (long long)N_NODES * NHEADS) rowsum[i] = 0.f;
}

// ---------------------------------------------------------------------------
// Kernel 3: per (node, head) scalars  s = x_h . wl,  t = y_h . wr
// ---------------------------------------------------------------------------
__global__ void gat_st(const float* __restrict__ x, const float* __restrict__ y,
                       const float* __restrict__ wl, const float* __restrict__ wr,
                       float* __restrict__ s, float* __restrict__ t) {
    int tid = blockIdx.x * blockDim.x + threadIdx.x;
    if (tid >= N_NODES * NHEADS) return;
    int node = tid >> 2, head = tid & 3;
    const float* xr = x + (size_t)node * NFEAT + head * DH;
    const float* yr = y + (size_t)node * NFEAT + head * DH;
    float ss = 0.f, tt = 0.f;
    #pragma unroll 8
    for (int k = 0; k < DH; ++k) {
        ss += xr[k] * wl[k];
        tt += yr[k] * wr[k];
    }
    s[tid] = ss;
    t[tid] = tt;
}

// ---------------------------------------------------------------------------
// Kernel 4: hy[n, head*64 + :] = y[n, head*64 + :] @ W  via V_WMMA_F32_16X16X4_F32
// One wave32 computes one (16-node tile, head) -> 16x64 f32 output tile.
// A (16x4 f32): lanes 0-15 hold M=0..15, v0/v1 = K,K+1; lanes 16-31 = K+2,K+3.
// B (4x16 f32): mirrored K striping, N = lane&15.  C/D (16x16 f32): 8 VGPRs.
// W is staged in LDS PRE-SWIZZLED in (K-pair, N) interleave so each lane's
// B operand {W[k][n], W[k+1][n]} is one aligned ds_load_b64 (no mov shuffles):
//   Wp[(k>>1)*128 + n*2 + (k&1)] = W[k][n]
// ---------------------------------------------------------------------------
__global__ __launch_bounds__(256) void gat_gemm_wmma(const float* __restrict__ y,
                                                     const float* __restrict__ W,
                                                     float* __restrict__ hy) {
    __shared__ float Wp[DH * DH];                       // 16 KB of 320 KB/WGP
    for (int i = threadIdx.x; i < DH * DH; i += 256) {
        const int k = i >> 6, n = i & 63;
        Wp[(k >> 1) * 128 + n * 2 + (k & 1)] = W[i];
    }
    __syncthreads();

    const int lane = threadIdx.x & 31;
    const int wave = threadIdx.x >> 5;
    const int pair = blockIdx.x * 8 + wave;             // (tile, head) pairs
    if (pair >= (N_NODES / 16) * NHEADS) return;        // wave-uniform exit
    const int tile  = pair >> 2;
    const int head  = pair & 3;
    const int node0 = tile * 16;
    const int m  = lane & 15;                           // row within tile / col within N-subtile
    const int hi = lane >> 4;                           // half-wave select

    const float* Arow = y + (size_t)(node0 + m) * NFEAT + head * DH;

    v8f acc[4] = {};                                    // 16x64 C in 32 VGPRs
    #pragma unroll
    for (int kb = 0; kb < DH; kb += 4) {
        const int ka = kb + hi * 2;                     // even -> 8B aligned
        const v2f a = *(const v2f*)(Arow + ka);         // global_load_b64
        const float* brow = &Wp[(ka >> 1) * 128 + m * 2];
        #pragma unroll
        for (int nt = 0; nt < 4; ++nt) {
            const v2f b = *(const v2f*)(brow + nt * 32);// ds_load_b64, contiguous pair
            acc[nt] = __builtin_amdgcn_wmma_f32_16x16x4_f32(
                false, a, false, b, (short)0, acc[nt], false, false);
        }
    }

    float* Crow = hy + (size_t)node0 * NFEAT + head * DH;
    #pragma unroll
    for (int nt = 0; nt < 4; ++nt) {
        #pragma unroll
        for (int r = 0; r < 8; ++r) {
            const int mm = r + hi * 8;                  // C layout: vgpr r, halves M=r / M=r+8
            const int nn = nt * 16 + m;
            Crow[(size_t)mm * NFEAT + nn] = acc[nt][r];
        }
    }
}

// ---------------------------------------------------------------------------
// Kernel 5: edge scatter. One wave32 per edge; lane covers 8 of 256 features,
// head = lane>>3 is lane-constant. Atomic f32 adds resolve at L2.
// ---------------------------------------------------------------------------
__global__ __launch_bounds__(256) void gat_edge(const int* __restrict__ ei,
                                                const float* __restrict__ s,
                                                const float* __restrict__ t,
                                                const float* __restrict__ hy,
                                                float* __restrict__ rowsum,
                                                float* __restrict__ out) {
    const int lane = threadIdx.x & 31;
    const long long e = (long long)blockIdx.x * 8 + (threadIdx.x >> 5);
    if (e >= N_EDGES) return;

    const int src = ei[e];
    const int dst = ei[N_EDGES + e];
    const int head = lane >> 3;

    const float* hrow = hy + (size_t)dst * NFEAT;
    const int base = lane * 8;
    __builtin_prefetch(hrow + base, 0, 0);              // global_prefetch_b8

    const float z   = s[src * NHEADS + head] + t[dst * NHEADS + head];
    const float lr  = z > 0.f ? z : ALPHA * z;          // leaky_relu
    const float att = expf(-lr);

    float* orow = out + (size_t)src * NFEAT;
    #pragma unroll
    for (int j = 0; j < 8; ++j)
        atomicAdd(orow + base + j, att * hrow[base + j]);   // global_atomic_add_f32
    if ((lane & 7) == 0)
        atomicAdd(rowsum + src * NHEADS + head, att);
}

// ---------------------------------------------------------------------------
// Kernel 6: normalize: out = rowsum>0 ? agg/rowsum : 0
// ---------------------------------------------------------------------------
__global__ void gat_final(float* __restrict__ out, const float* __restrict__ rowsum) {
    long long i = (long long)blockIdx.x * 256 + threadIdx.x;
    if (i >= (long long)N_NODES * NFEAT) return;
    const int node = (int)(i >> 8);
    const int head = (int)((i >> 6) & 3);
    const float rs = rowsum[node * NHEADS + head];
    const float v  = out[i];
    out[i] = rs > 0.f ? v / rs : 0.f;
}

// ---------------------------------------------------------------------------
extern "C" void kernel_launch(void* const* d_in, const int* in_sizes, int n_in,
                              void* d_out, int out_size, void* d_ws, size_t ws_size,
                              hipStream_t stream) {
    const float* x  = (const float*)d_in[0];
    const float* y  = (const float*)d_in[1];
    const float* W  = (const float*)d_in[2];
    const float* a  = (const float*)d_in[3];
    const int*   ei = (const int*)d_in[4];
    float* out = (float*)d_out;

    // workspace layout (floats)
    float* ws = (float*)d_ws;
    float* wl     = ws;                                   // 64
    float* wr     = ws + 64;                              // 64
    float* s      = ws + 128;                             // 200000
    float* t      = s + (size_t)N_NODES * NHEADS;         // 200000
    float* hy     = t + (size_t)N_NODES * NHEADS;         // 12.8M
    float* rowsum = hy + (size_t)N_NODES * NFEAT;         // 200000

    gat_prep<<<1, 64, 0, stream>>>(W, a, wl, wr);

    const int total_out = N_NODES * NFEAT;                // 12,800,000
    gat_zero<<<(total_out + 255) / 256, 256, 0, stream>>>(out, rowsum);

    gat_st<<<(N_NODES * NHEADS + 255) / 256, 256, 0, stream>>>(x, y, wl, wr, s, t);

    const int pairs = (N_NODES / 16) * NHEADS;            // 12,500 waves
    gat_gemm_wmma<<<(pairs + 7) / 8, 256, 0, stream>>>(y, W, hy);

    gat_edge<<<(N_EDGES + 7) / 8, 256, 0, stream>>>(ei, s, t, hy, rowsum, out);

    gat_final<<<(total_out + 255) / 256, 256, 0, stream>>>(out, rowsum);
}